// VectorQuantizer_30872224923700
// MI455X (gfx1250) — compile-verified
//
#include <hip/hip_runtime.h>
#include <hip/hip_bf16.h>
#include <math.h>

// ---- problem constants (from reference) ----
#define NROWS   65536          // 64*32*32
#define EMB_DIM 256            // D
#define KCODES  1024           // number of embeddings
#define BETA    0.25f
#define MROWS   32             // rows per block (2 WMMA M-tiles share each B fragment)

// output layout (tuple concatenated flat, all as float):
#define OUT_Q_OFF    0
#define OUT_IDX_OFF  16777216
#define OUT_VQ_OFF   16842752

// workspace layout (floats): enorm[1024], counts[1024], losssum[1]
#define WS_ENORM  0
#define WS_COUNTS 1024
#define WS_LOSS   2048

typedef __attribute__((ext_vector_type(2))) float v2f;
typedef __attribute__((ext_vector_type(8))) float v8f;

// LDS z tile padded row stride: 260 % 64 == 4 -> 16 rows per half map to
// distinct bank groups {4r..4r+3}: conflict-free b64 A-fragment reads.
#define ZT_STRIDE 260

// ---------------- kernel 0: zero accumulators ----------------
__global__ void vq_init_kernel(float* __restrict__ counts, float* __restrict__ losssum) {
    int i = blockIdx.x * 256 + threadIdx.x;
    if (i < KCODES) counts[i] = 0.0f;
    if (i == 0) *losssum = 0.0f;
}

// ---------------- kernel 1: ||e_k||^2 per code (coalesced over k) ----------------
__global__ void vq_enorm_kernel(const float* __restrict__ E, float* __restrict__ enorm) {
    int k = blockIdx.x * 256 + threadIdx.x;   // 4 blocks x 256 = 1024
    float s = 0.0f;
    #pragma unroll 8
    for (int d = 0; d < EMB_DIM; ++d) {
        float e = E[d * KCODES + k];
        s = fmaf(e, e, s);
    }
    enorm[k] = s;
}

// ---------------- kernel 2: fused WMMA GEMM + argmin + quantize + loss ----------------
// One block = 32 rows of z. 8 waves; wave w covers codes [w*128, w*128+128) in
// eight 16x16 N-tiles. Per K-step: one B fragment (2 global b32) feeds TWO
// V_WMMA_F32_16X16X4_F32 (rows 0-15 and rows 16-31) -> 1 load-pair : 2 WMMA.
__global__ __launch_bounds__(256) void vq_main_kernel(
        const float* __restrict__ z, const float* __restrict__ E,
        const float* __restrict__ enorm,
        float* __restrict__ outq, float* __restrict__ outidx,
        float* __restrict__ counts, float* __restrict__ losssum) {

    __shared__ float zt[MROWS * ZT_STRIDE];   // 32x256 z tile, padded (~33 KB)
    __shared__ float cw_val[MROWS * 8];       // per-(row, wave) argmin candidates
    __shared__ int   cw_idx[MROWS * 8];
    __shared__ int   fin[MROWS];              // final index per row
    __shared__ float rsum[256];               // loss reduction

    const int t  = threadIdx.x;
    const int m0 = blockIdx.x * MROWS;

    // ---- stage 32x256 z tile into LDS (coalesced float4) ----
    {
        const float* zg = z + (size_t)m0 * EMB_DIM;
        #pragma unroll
        for (int j = 0; j < 8; ++j) {
            int fi = j * 1024 + t * 4;                // flat float index in tile
            float4 v = *(const float4*)(zg + fi);
            int r = fi >> 8;                          // /256
            int c = fi & 255;
            *(float4*)&zt[r * ZT_STRIDE + c] = v;     // 16B aligned (1040 & 15 == 0)
        }
    }
    __syncthreads();

    const int wave = t >> 5;
    const int lane = t & 31;
    const int row  = lane & 15;       // N-column track within half-wave
    const int hi   = lane >> 4;       // 0: K+0/1, 1: K+2/3

    float bestv0[8], bestv1[8];
    int   besti0[8], besti1[8];
    #pragma unroll
    for (int v = 0; v < 8; ++v) {
        bestv0[v] = 3.4e38f; besti0[v] = 0;
        bestv1[v] = 3.4e38f; besti1[v] = 0;
    }

    const float* ap0 = &zt[row * ZT_STRIDE + hi * 2];           // rows 0-15 tile
    const float* ap1 = ap0 + 16 * ZT_STRIDE;                    // rows 16-31 tile

    for (int nt = 0; nt < 8; ++nt) {
        const int n0 = wave * 128 + nt * 16;
        const int n  = n0 + row;                       // this lane's code column
        const float* bp = E + (size_t)(hi * 2) * KCODES + n;

        v8f c0 = {0.f, 0.f, 0.f, 0.f, 0.f, 0.f, 0.f, 0.f};
        v8f c1 = {0.f, 0.f, 0.f, 0.f, 0.f, 0.f, 0.f, 0.f};
        #pragma unroll 16
        for (int k0 = 0; k0 < EMB_DIM; k0 += 4) {
            v2f b;
            b.x = bp[(size_t)k0 * KCODES];             // E[(k0+hi*2)  ][n]
            b.y = bp[(size_t)(k0 + 1) * KCODES];       // E[(k0+hi*2+1)][n]
            float2 a0f = *(const float2*)(ap0 + k0);   // ds_load_b64, conflict-free
            float2 a1f = *(const float2*)(ap1 + k0);
            v2f a0; a0.x = a0f.x; a0.y = a0f.y;
            v2f a1; a1.x = a1f.x; a1.y = a1f.y;
            c0 = __builtin_amdgcn_wmma_f32_16x16x4_f32(
                    false, a0, false, b, (short)0, c0, false, false);
            c1 = __builtin_amdgcn_wmma_f32_16x16x4_f32(
                    false, a1, false, b, (short)0, c1, false, false);
        }
        const float en = enorm[n];
        #pragma unroll
        for (int v = 0; v < 8; ++v) {
            float s0 = en - 2.0f * c0[v];              // ||e||^2 - 2 z.e
            if (s0 < bestv0[v]) { bestv0[v] = s0; besti0[v] = n; }
            float s1 = en - 2.0f * c1[v];
            if (s1 < bestv1[v]) { bestv1[v] = s1; besti1[v] = n; }
        }
    }

    // ---- in-wave butterfly argmin across the 16 column-tracks (lanes sharing hi) ----
    #pragma unroll
    for (int v = 0; v < 8; ++v) {
        float bv0 = bestv0[v]; int bi0 = besti0[v];
        float bv1 = bestv1[v]; int bi1 = besti1[v];
        #pragma unroll
        for (int off = 8; off > 0; off >>= 1) {        // masks 1,2,4,8 stay in half
            float ov0 = __shfl_xor(bv0, off, 32);
            int   oi0 = __shfl_xor(bi0, off, 32);
            if (ov0 < bv0 || (ov0 == bv0 && oi0 < bi0)) { bv0 = ov0; bi0 = oi0; }
            float ov1 = __shfl_xor(bv1, off, 32);
            int   oi1 = __shfl_xor(bi1, off, 32);
            if (ov1 < bv1 || (ov1 == bv1 && oi1 < bi1)) { bv1 = ov1; bi1 = oi1; }
        }
        bestv0[v] = bv0; besti0[v] = bi0;
        bestv1[v] = bv1; besti1[v] = bi1;
    }

    // lanes 0 and 16 now hold the wave's best per row; deposit 8 waves x 32 rows
    if (row == 0) {
        #pragma unroll
        for (int v = 0; v < 8; ++v) {
            int r0 = v + hi * 8;                       // rows 0-15 (c0 set)
            cw_val[r0 * 8 + wave] = bestv0[v];
            cw_idx[r0 * 8 + wave] = besti0[v];
            int r1 = r0 + 16;                          // rows 16-31 (c1 set)
            cw_val[r1 * 8 + wave] = bestv1[v];
            cw_idx[r1 * 8 + wave] = besti1[v];
        }
    }
    __syncthreads();

    // ---- final 8 -> 1 per row; emit index + histogram ----
    if (t < MROWS) {
        int r = t;
        float bv = cw_val[r * 8];
        int   bi = cw_idx[r * 8];
        #pragma unroll
        for (int j = 1; j < 8; ++j) {
            float v2 = cw_val[r * 8 + j];
            int   i2 = cw_idx[r * 8 + j];
            if (v2 < bv || (v2 == bv && i2 < bi)) { bv = v2; bi = i2; }
        }
        fin[r] = bi;
        outidx[m0 + r] = (float)bi;
        atomicAdd(&counts[bi], 1.0f);
    }
    __syncthreads();

    // ---- quantize (gather e[:,idx]) + squared-error partial sum ----
    float lsq = 0.0f;
    #pragma unroll 4
    for (int r = 0; r < MROWS; ++r) {
        int bi  = fin[r];
        float q = E[(size_t)t * KCODES + bi];          // e^T row gather (L2-resident)
        float zv = zt[r * ZT_STRIDE + t];
        float d = q - zv;
        lsq = fmaf(d, d, lsq);
        outq[(size_t)(m0 + r) * EMB_DIM + t] = q;      // quantized_st == quantized
    }
    rsum[t] = lsq;
    __syncthreads();
    for (int s = 128; s > 0; s >>= 1) {
        if (t < s) rsum[t] += rsum[t + s];
        __syncthreads();
    }
    if (t == 0) atomicAdd(losssum, rsum[0]);
}

// ---------------- kernel 3: perplexity + scalar losses ----------------
__global__ __launch_bounds__(1024) void vq_final_kernel(
        const float* __restrict__ counts, const float* __restrict__ losssum,
        float* __restrict__ out) {
    __shared__ float sh[1024];
    int t = threadIdx.x;
    float p = counts[t] * (1.0f / (float)NROWS);
    sh[t] = p * logf(p + 1e-10f);
    __syncthreads();
    for (int s = 512; s > 0; s >>= 1) {
        if (t < s) sh[t] += sh[t + s];
        __syncthreads();
    }
    if (t == 0) {
        float vq = *losssum * (1.0f / (float)((size_t)NROWS * EMB_DIM));
        out[OUT_VQ_OFF + 0] = vq;
        out[OUT_VQ_OFF + 1] = BETA * vq;   // commitment = beta * same MSE
        out[OUT_VQ_OFF + 2] = expf(-sh[0]);
    }
}

extern "C" void kernel_launch(void* const* d_in, const int* in_sizes, int n_in,
                              void* d_out, int out_size, void* d_ws, size_t ws_size,
                              hipStream_t stream) {
    const float* z = (const float*)d_in[0];   // [65536, 256] flattened
    const float* E = (const float*)d_in[1];   // [256, 1024]
    float* out = (float*)d_out;
    float* ws  = (float*)d_ws;

    float* enorm   = ws + WS_ENORM;
    float* counts  = ws + WS_COUNTS;
    float* losssum = ws + WS_LOSS;

    vq_init_kernel <<<4, 256, 0, stream>>>(counts, losssum);
    vq_enorm_kernel<<<KCODES / 256, 256, 0, stream>>>(E, enorm);
    vq_main_kernel <<<NROWS / MROWS, 256, 0, stream>>>(
        z, E, enorm, out + OUT_Q_OFF, out + OUT_IDX_OFF, counts, losssum);
    vq_final_kernel<<<1, 1024, 0, stream>>>(counts, losssum, out);
}